// RGCN_11553462026387
// MI455X (gfx1250) — compile-verified
//
#include <hip/hip_runtime.h>

typedef __attribute__((ext_vector_type(2))) float v2f;
typedef __attribute__((ext_vector_type(8))) float v8f;

#define N_REL 4

// ---------------------------------------------------------------------------
// Dense transform with CDNA5 fp32 WMMA (V_WMMA_F32_16X16X4_F32).
//   rel in [0..3]: xWbuf[rel] = h @ W[rel]          ([N, DOUT])
//   rel == 4:      outbuf     = h @ root + bias     ([N, DOUT])
// One wave handles a 16-node tile for one relation. W tile staged in LDS.
// DIN/DOUT are compile-time so the k/nt loops fully unroll and the v8f
// accumulators stay in statically-indexed VGPRs.
// A frag (16x4 f32): lanes 0-15 -> K={0,1}, lanes 16-31 -> K={2,3},
//   M = lane&15  -> one contiguous 8B global load per lane.
// B frag (4x16 f32): same K split, Ncol = lane&15 -> ds_load_2addr pair.
// C/D (16x16 f32, 8 VGPRs): M = j + 8*(lane>=16), Ncol = lane&15.
// Epilogue: wave-uniform full-tile fast path -> unguarded immediate-offset
// stores (no per-element EXEC manipulation); ragged tail falls back to the
// guarded path.
// ---------------------------------------------------------------------------
template <int DIN, int DOUT>
__global__ __launch_bounds__(128)
void rgcn_dense_wmma(const float* __restrict__ h,     // [N, DIN]
                     const float* __restrict__ W,     // [R, DIN, DOUT]
                     const float* __restrict__ root,  // [DIN, DOUT]
                     const float* __restrict__ bias,  // [DOUT]
                     float* __restrict__ xWbuf,       // [R, N, DOUT]
                     float* __restrict__ outbuf,      // [N, DOUT]
                     int N)
{
    __shared__ float sW[DIN * DOUT];
    const int rel = blockIdx.y;                       // 0..4 (4 == root)
    const float* Wp = (rel < N_REL) ? (W + (size_t)rel * DIN * DOUT) : root;
    for (int i = threadIdx.x; i < DIN * DOUT; i += 128)
        sW[i] = Wp[i];
    __syncthreads();

    const int wave   = threadIdx.x >> 5;
    const int lane   = threadIdx.x & 31;
    const int ntiles = (N + 15) >> 4;
    const int tile   = blockIdx.x * 4 + wave;         // 4 waves per block
    if (tile >= ntiles) return;                       // wave-uniform: EXEC all-1s

    const int m     = lane & 15;                      // M row (A) / N col (B,D)
    const int khalf = lane >> 4;                      // 0 -> K pair {0,1}, 1 -> {2,3}
    constexpr int NT = DOUT / 16;                     // 1, 2 or 4 column tiles

    v8f acc[NT];
#pragma unroll
    for (int nt = 0; nt < NT; ++nt)
        acc[nt] = (v8f){0.f, 0.f, 0.f, 0.f, 0.f, 0.f, 0.f, 0.f};

    int row = tile * 16 + m;
    if (row >= N) row = N - 1;
    const float* hrow = h + (size_t)row * DIN;

#pragma unroll
    for (int k0 = 0; k0 < DIN; k0 += 4) {
        const v2f a = *(const v2f*)(hrow + k0 + khalf * 2);   // global_load_b64
#pragma unroll
        for (int nt = 0; nt < NT; ++nt) {
            const float* wp0 = &sW[(k0 + khalf * 2) * DOUT + nt * 16 + m];
            v2f b;
            b.x = wp0[0];
            b.y = wp0[DOUT];
            acc[nt] = __builtin_amdgcn_wmma_f32_16x16x4_f32(
                false, a, false, b, (short)0, acc[nt], false, false);
        }
    }

    // ---- epilogue -------------------------------------------------------
    const int r0 = tile * 16 + khalf * 8;             // first of 8 rows per lane
    const bool fulltile = (tile * 16 + 16 <= N);      // wave-uniform

    if (rel < N_REL) {
        float* base = xWbuf + (size_t)rel * N * DOUT + (size_t)r0 * DOUT + m;
        if (fulltile) {
#pragma unroll
            for (int nt = 0; nt < NT; ++nt)
#pragma unroll
                for (int j = 0; j < 8; ++j)
                    base[(size_t)j * DOUT + nt * 16] = acc[nt][j];
        } else {
#pragma unroll
            for (int nt = 0; nt < NT; ++nt)
#pragma unroll
                for (int j = 0; j < 8; ++j)
                    if (r0 + j < N)
                        base[(size_t)j * DOUT + nt * 16] = acc[nt][j];
        }
    } else {
        float* base = outbuf + (size_t)r0 * DOUT + m;
        float bv[NT];
#pragma unroll
        for (int nt = 0; nt < NT; ++nt)
            bv[nt] = bias[nt * 16 + m];
        if (fulltile) {
#pragma unroll
            for (int nt = 0; nt < NT; ++nt)
#pragma unroll
                for (int j = 0; j < 8; ++j)
                    base[(size_t)j * DOUT + nt * 16] = acc[nt][j] + bv[nt];
        } else {
#pragma unroll
            for (int nt = 0; nt < NT; ++nt)
#pragma unroll
                for (int j = 0; j < 8; ++j)
                    if (r0 + j < N)
                        base[(size_t)j * DOUT + nt * 16] = acc[nt][j] + bv[nt];
        }
    }
}

// ---------------------------------------------------------------------------
// Edge-degree counting per (relation, dst) segment — shared across all layers.
// fp32 counter is exact up to 2^24 (degrees here are << that).
// ---------------------------------------------------------------------------
__global__ void rgcn_count(const int* __restrict__ et, const int* __restrict__ dst,
                           float* __restrict__ cnt, int E, int N)
{
    int e = blockIdx.x * blockDim.x + threadIdx.x;
    if (e < E)
        atomicAdd(&cnt[(size_t)et[e] * N + dst[e]], 1.0f);
}

__global__ void rgcn_invert(float* __restrict__ cnt, int n)
{
    int i = blockIdx.x * blockDim.x + threadIdx.x;
    if (i < n)
        cnt[i] = 1.0f / fmaxf(cnt[i], 1.0f);
}

__global__ void rgcn_zero(float4* __restrict__ p, long long n4)
{
    long long i = (long long)blockIdx.x * blockDim.x + threadIdx.x;
    if (i < n4)
        p[i] = make_float4(0.f, 0.f, 0.f, 0.f);
}

// ---------------------------------------------------------------------------
// Scatter: group of (dout/4) threads per edge; each thread moves a float4 of
// xW[rel, src] into sums[rel, dst] via 4x global_atomic_add_f32. The sums
// buffer (<=102 MB) fits in the 192 MB L2, so atomics resolve on-chip.
// ---------------------------------------------------------------------------
__global__ void rgcn_scatter(const float* __restrict__ xW,
                             const int* __restrict__ src, const int* __restrict__ dst,
                             const int* __restrict__ et,
                             float* __restrict__ sums,
                             long long total, int N, int dout, int ldq)
{
    long long t = (long long)blockIdx.x * blockDim.x + threadIdx.x;
    if (t >= total) return;
    const int e = (int)(t >> ldq);
    const int f = ((int)t & ((1 << ldq) - 1)) << 2;
    const int r = et[e];
    const int s = src[e];
    const int d = dst[e];
    const float4 v = *(const float4*)(xW + ((size_t)r * N + s) * dout + f);
    float* p = sums + ((size_t)r * N + d) * dout + f;
    atomicAdd(p + 0, v.x);
    atomicAdd(p + 1, v.y);
    atomicAdd(p + 2, v.z);
    atomicAdd(p + 3, v.w);
}

// ---------------------------------------------------------------------------
// Finalize: out[n,:] += sum_r sums[r,n,:] * inv_cnt[r,n]; optional fused ReLU.
// out already holds h@root + bias from the dense kernel.
// ---------------------------------------------------------------------------
__global__ void rgcn_finalize(const float* __restrict__ sums,
                              const float* __restrict__ inv,
                              float* __restrict__ out,
                              int N, int dout, int ldq, int relu)
{
    long long t = (long long)blockIdx.x * blockDim.x + threadIdx.x;
    long long total = ((long long)N * dout) >> 2;
    if (t >= total) return;
    const int n = (int)(t >> ldq);
    const int f = ((int)t & ((1 << ldq) - 1)) << 2;
    float4 a = *(const float4*)(out + (size_t)n * dout + f);
#pragma unroll
    for (int r = 0; r < N_REL; ++r) {
        const float iv = inv[(size_t)r * N + n];
        const float4 s = *(const float4*)(sums + ((size_t)r * N + n) * dout + f);
        a.x += s.x * iv;
        a.y += s.y * iv;
        a.z += s.z * iv;
        a.w += s.w * iv;
    }
    if (relu) {
        a.x = fmaxf(a.x, 0.f);
        a.y = fmaxf(a.y, 0.f);
        a.z = fmaxf(a.z, 0.f);
        a.w = fmaxf(a.w, 0.f);
    }
    *(float4*)(out + (size_t)n * dout + f) = a;
}

// ---------------------------------------------------------------------------
static inline int ilog2i(int x) { int l = 0; while ((1 << l) < x) ++l; return l; }

template <int DIN, int DOUT>
static void run_layer(const float* h, const float* W, const float* root, const float* bias,
                      float* xW, float* sums, const float* inv, float* out,
                      const int* src, const int* dst, const int* et,
                      int N, int E, int relu, hipStream_t stream)
{
    const int ntiles = (N + 15) >> 4;
    dim3 dgrid((ntiles + 3) / 4, N_REL + 1);
    rgcn_dense_wmma<DIN, DOUT><<<dgrid, 128, 0, stream>>>(h, W, root, bias, xW, out, N);

    const long long sn4 = ((long long)N_REL * N * DOUT) >> 2;
    rgcn_zero<<<(unsigned)((sn4 + 255) / 256), 256, 0, stream>>>((float4*)sums, sn4);

    const int ldq = ilog2i(DOUT >> 2);
    const long long total = (long long)E << ldq;
    rgcn_scatter<<<(unsigned)((total + 255) / 256), 256, 0, stream>>>(
        xW, src, dst, et, sums, total, N, DOUT, ldq);

    const long long ft = ((long long)N * DOUT) >> 2;
    rgcn_finalize<<<(unsigned)((ft + 255) / 256), 256, 0, stream>>>(
        sums, inv, out, N, DOUT, ldq, relu);
}

extern "C" void kernel_launch(void* const* d_in, const int* in_sizes, int n_in,
                              void* d_out, int out_size, void* d_ws, size_t ws_size,
                              hipStream_t stream)
{
    const float* x  = (const float*)d_in[0];   // [N, 16]
    const int*   ei = (const int*)d_in[1];     // [2, E]
    const int*   et = (const int*)d_in[2];     // [E]
    const float* W1 = (const float*)d_in[3];
    const float* r1 = (const float*)d_in[4];
    const float* b1 = (const float*)d_in[5];
    const float* W2 = (const float*)d_in[6];
    const float* r2 = (const float*)d_in[7];
    const float* b2 = (const float*)d_in[8];
    const float* W3 = (const float*)d_in[9];
    const float* r3 = (const float*)d_in[10];
    const float* b3 = (const float*)d_in[11];

    const int N = in_sizes[0] / 16;            // 100000
    const int E = in_sizes[2];                 // 1600000
    const int* src = ei;
    const int* dst = ei + E;

    float* ws = (float*)d_ws;
    size_t p = 0;
    float* xW   = ws + p; p += (size_t)N_REL * N * 64;   // sized for max dout
    float* sums = ws + p; p += (size_t)N_REL * N * 64;
    float* inv  = ws + p; p += (size_t)N_REL * N;
    float* h1   = ws + p; p += (size_t)N * 16;
    float* h2   = ws + p; p += (size_t)N * 32;
    float* out  = (float*)d_out;               // [N, 64]

    // Degree counts per (relation, dst) are identical across the 3 layers:
    // compute once, invert once, reuse.
    const long long cn  = (long long)N_REL * N;
    const long long cn4 = cn >> 2;
    rgcn_zero<<<(unsigned)((cn4 + 255) / 256), 256, 0, stream>>>((float4*)inv, cn4);
    rgcn_count<<<(E + 255) / 256, 256, 0, stream>>>(et, dst, inv, E, N);
    rgcn_invert<<<(unsigned)((cn + 255) / 256), 256, 0, stream>>>(inv, (int)cn);

    run_layer<16, 16>(x,  W1, r1, b1, xW, sums, inv, h1,  src, dst, et, N, E, 1, stream);
    run_layer<16, 32>(h1, W2, r2, b2, xW, sums, inv, h2,  src, dst, et, N, E, 1, stream);
    run_layer<32, 64>(h2, W3, r3, b3, xW, sums, inv, out, src, dst, et, N, E, 0, stream);
}